// ProtoNet_86870008529163
// MI455X (gfx1250) — compile-verified
//
#include <hip/hip_runtime.h>
#include <hip/hip_bf16.h>

#define WAY   5
#define SHOT  5
#define QUERY 15
#define EMB   640
#define NMEM  100000
#define NSUP  (WAY*SHOT)          // 25
#define NQ    (WAY*QUERY)         // 75
#define NCAND (SHOT + NMEM)       // 100005 candidates per way
#define TOPK  8
#define EPSN  1e-12f

typedef float v2f __attribute__((ext_vector_type(2)));
typedef float v8f __attribute__((ext_vector_type(8)));

__device__ __forceinline__ float waveReduceSum(float v) {
    v += __shfl_xor(v, 16, 32);
    v += __shfl_xor(v, 8, 32);
    v += __shfl_xor(v, 4, 32);
    v += __shfl_xor(v, 2, 32);
    v += __shfl_xor(v, 1, 32);
    return v;
}

__device__ __forceinline__ float dot4(float4 a, float4 b) {
    return a.x*b.x + a.y*b.y + a.z*b.z + a.w*b.w;
}

// ---------------- K1: support norms, mean-of-normalized vectors, sim_sup -------------
__global__ __launch_bounds__(256)
void k1_prep(const float* __restrict__ emb, float* __restrict__ mv,
             float* __restrict__ simSup) {
    __shared__ float invn[NSUP];
    __shared__ float mvLds[WAY*EMB];   // 12.8 KB
    const int tid = threadIdx.x, lane = tid & 31, wv = tid >> 5;

    // Phase A: inverse norms of the 25 support rows (wave per row)
    for (int r = wv; r < NSUP; r += 8) {
        const float* row = emb + r*EMB;
        float ss = 0.f;
        for (int j = lane; j < EMB; j += 32) { float x = row[j]; ss += x*x; }
        ss = waveReduceSum(ss);
        if (lane == 0) invn[r] = 1.0f / fmaxf(sqrtf(ss), EPSN);
    }
    __syncthreads();

    // Phase B: mv[w][d] = mean_s sup_n[w][s][d]   (sup row index = s*WAY + w)
    for (int d = tid; d < EMB; d += 256) {
        #pragma unroll
        for (int w = 0; w < WAY; ++w) {
            float acc = 0.f;
            #pragma unroll
            for (int s = 0; s < SHOT; ++s) {
                int r = s*WAY + w;
                acc += emb[r*EMB + d] * invn[r];
            }
            acc *= (1.0f / SHOT);
            mvLds[w*EMB + d] = acc;
            mv[w*EMB + d]    = acc;
        }
    }
    __syncthreads();

    // Phase C: sim_sup[w][t] = mv[w] . sup_n[w][t]
    for (int p = wv; p < NSUP; p += 8) {
        int w = p / WAY, t = p % WAY;
        int r = t*WAY + w;
        const float* row = emb + r*EMB;
        float acc = 0.f;
        for (int j = lane; j < EMB; j += 32) acc += mvLds[w*EMB + j] * row[j];
        acc = waveReduceSum(acc);
        if (lane == 0) simSup[w*WAY + t] = acc * invn[r];
    }
}

// ---------------- K2: the bandwidth-bound pass over the 256 MB memory bank ----------
// One wave per row; lane-striped float4 loads (32 lanes x 16B = 512B contiguous per
// step); mean vectors held in 100 VGPRs per lane; 6 cross-lane reductions per row.
__global__ __launch_bounds__(256)
void k2_simmem(const float* __restrict__ mem, const float* __restrict__ mv,
               float* __restrict__ simMem) {
    const int lane   = threadIdx.x & 31;
    const int wvGlob = blockIdx.x * (blockDim.x >> 5) + (threadIdx.x >> 5);
    const int nWaves = gridDim.x * (blockDim.x >> 5);

    // Pin mv in registers: mvr[w][j] = mv[w] at float4 index (lane + 32*j)
    float4 mvr[WAY][5];
    #pragma unroll
    for (int w = 0; w < WAY; ++w) {
        const float4* p = (const float4*)(mv + w*EMB);
        #pragma unroll
        for (int j = 0; j < 5; ++j) mvr[w][j] = p[lane + 32*j];
    }

    for (int row = wvGlob; row < NMEM; row += nWaves) {
        const float4* rp = (const float4*)(mem + (size_t)row * EMB);
        int nrow = row + nWaves;
        if (nrow < NMEM && lane < 20)   // 20 x 128B lines cover the next 2560B row
            __builtin_prefetch(mem + (size_t)nrow * EMB + lane*32, 0, 1);

        float nrm = 0.f, a0 = 0.f, a1 = 0.f, a2 = 0.f, a3 = 0.f, a4 = 0.f;
        #pragma unroll
        for (int j = 0; j < 5; ++j) {
            float4 v = rp[lane + 32*j];
            nrm += dot4(v, v);
            a0 += dot4(v, mvr[0][j]);
            a1 += dot4(v, mvr[1][j]);
            a2 += dot4(v, mvr[2][j]);
            a3 += dot4(v, mvr[3][j]);
            a4 += dot4(v, mvr[4][j]);
        }
        nrm = waveReduceSum(nrm);
        a0 = waveReduceSum(a0);
        a1 = waveReduceSum(a1);
        a2 = waveReduceSum(a2);
        a3 = waveReduceSum(a3);
        a4 = waveReduceSum(a4);
        float inv = 1.0f / fmaxf(sqrtf(nrm), EPSN);
        float simv = (lane == 0) ? a0 : (lane == 1) ? a1 : (lane == 2) ? a2
                   : (lane == 3) ? a3 : a4;
        if (lane < WAY) simMem[lane * NMEM + row] = simv * inv;
    }
}

// ---------------- K3: exact per-way top-8 over 100005 candidates --------------------
__global__ __launch_bounds__(1024)
void k3_topk(const float* __restrict__ simSup, const float* __restrict__ simMem,
             float* __restrict__ topVals, int* __restrict__ topIdx) {
    const int w = blockIdx.x;
    const int tid = threadIdx.x;
    const int NT = 1024;
    __shared__ float sval[NT];
    __shared__ int   sidx[NT];
    __shared__ int   sthr[NT];

    // Per-thread register top-8 (sorted descending)
    float lv[TOPK]; int li[TOPK];
    #pragma unroll
    for (int k = 0; k < TOPK; ++k) { lv[k] = -1e30f; li[k] = 0; }

    for (int c = tid; c < NCAND; c += NT) {
        float v = (c < SHOT) ? simSup[w*WAY + c] : simMem[w*NMEM + (c - SHOT)];
        if (v > lv[TOPK-1]) {
            lv[TOPK-1] = v; li[TOPK-1] = c;
            #pragma unroll
            for (int k = TOPK-1; k > 0; --k) {
                if (lv[k] > lv[k-1]) {
                    float tv = lv[k]; lv[k] = lv[k-1]; lv[k-1] = tv;
                    int   ti = li[k]; li[k] = li[k-1]; li[k-1] = ti;
                }
            }
        }
    }

    // 8 rounds of block-wide argmax; winner pops its local list
    int p = 0;
    for (int k = 0; k < TOPK; ++k) {
        sval[tid] = (p < TOPK) ? lv[p] : -1e30f;
        sidx[tid] = (p < TOPK) ? li[p] : 0;
        sthr[tid] = tid;
        __syncthreads();
        for (int s = NT/2; s > 0; s >>= 1) {
            if (tid < s && sval[tid + s] > sval[tid]) {
                sval[tid] = sval[tid + s];
                sidx[tid] = sidx[tid + s];
                sthr[tid] = sthr[tid + s];
            }
            __syncthreads();
        }
        if (tid == 0) { topVals[w*TOPK + k] = sval[0]; topIdx[w*TOPK + k] = sidx[0]; }
        int winner = sthr[0];
        __syncthreads();
        if (tid == winner) ++p;
    }
}

// ---------------- K4: prototype = weighted sum of RAW sources, then L2 normalize ----
__global__ __launch_bounds__(1024)
void k4_proto(const float* __restrict__ emb, const float* __restrict__ mem,
              const float* __restrict__ topVals, const int* __restrict__ topIdx,
              float* __restrict__ protoN) {
    __shared__ float red[1024];
    const int d = threadIdx.x;
    float pr[WAY];

    #pragma unroll
    for (int w = 0; w < WAY; ++w) {
        float num = 0.f, den = 0.f;
        #pragma unroll
        for (int k = 0; k < TOPK; ++k) {
            float v = topVals[w*TOPK + k];
            int   c = topIdx[w*TOPK + k];
            den += v;
            if (d < EMB) {
                const float* src = (c < SHOT) ? (emb + (c*WAY + w)*EMB)
                                              : (mem + (size_t)(c - SHOT)*EMB);
                num += v * src[d];
            }
        }
        pr[w] = (d < EMB) ? (num / den) : 0.f;
    }

    #pragma unroll
    for (int w = 0; w < WAY; ++w) {
        red[d] = pr[w]*pr[w];
        __syncthreads();
        for (int s = 512; s > 0; s >>= 1) {
            if (d < s) red[d] += red[d + s];
            __syncthreads();
        }
        float scale = 1.0f / fmaxf(sqrtf(red[0]), EPSN);
        __syncthreads();
        if (d < EMB) protoN[w*EMB + d] = pr[w] * scale;
    }
}

// ---------------- K5: logits = query @ proto_n^T / 64 via V_WMMA_F32_16X16X4_F32 ----
// A (16x4 f32): lane L: M = L&15, K-pair base = (L>>4)*2, 2 VGPRs (v2f).
// B (4x16 f32): lane L: N = L&15, same K-pair split.
// C/D (16x16 f32): VGPR v: lanes 0-15 -> M=v, N=lane; lanes 16-31 -> M=v+8, N=lane-16.
__global__ __launch_bounds__(32)
void k5_logits(const float* __restrict__ emb, const float* __restrict__ protoN,
               float* __restrict__ out) {
    const int tile = blockIdx.x;          // 5 tiles of 16 query rows (75 padded to 80)
    const int L = threadIdx.x;
    const int mrow = L & 15;
    const int kb = (L >> 4) * 2;
    const int n = L & 15;
    const bool nvalid = n < WAY;

    int q = tile*16 + mrow;
    int qq = q < NQ ? q : (NQ - 1);
    const float* arow = emb + (NSUP + qq) * EMB;          // query rows start at 25
    const float* brow = protoN + (nvalid ? n : 0) * EMB;  // B[k][n] = protoN[n][k]

    v8f c = {0.f, 0.f, 0.f, 0.f, 0.f, 0.f, 0.f, 0.f};
    for (int k0 = 0; k0 < EMB; k0 += 4) {
        v2f a, b;
        a.x = arow[k0 + kb];
        a.y = arow[k0 + kb + 1];
        float bx = brow[k0 + kb];
        float by = brow[k0 + kb + 1];
        b.x = nvalid ? bx : 0.f;
        b.y = nvalid ? by : 0.f;
        c = __builtin_amdgcn_wmma_f32_16x16x4_f32(false, a, false, b,
                                                  (short)0, c, false, false);
    }

    #pragma unroll
    for (int v = 0; v < 8; ++v) {
        int m = v + ((L >> 4) << 3);
        int qrow = tile*16 + m;
        if (qrow < NQ && n < WAY)
            out[qrow*WAY + n] = c[v] * (1.0f / 64.0f);
    }
}

// ---------------- launcher ----------------------------------------------------------
extern "C" void kernel_launch(void* const* d_in, const int* in_sizes, int n_in,
                              void* d_out, int out_size, void* d_ws, size_t ws_size,
                              hipStream_t stream) {
    const float* emb = (const float*)d_in[0];   // (100, 640) fp32
    const float* mem = (const float*)d_in[1];   // (100000, 640) fp32
    float* out = (float*)d_out;                 // (75, 5) fp32

    // workspace layout (floats): mv | simSup | simMem | topVals | topIdx | protoN
    float* ws      = (float*)d_ws;
    float* mv      = ws;                        // 3200
    float* simSup  = ws + 3200;                 // 25 (padded to 32)
    float* simMem  = ws + 3232;                 // 500000  (~2 MB)
    float* topVals = ws + 503232;               // 40
    int*   topIdx  = (int*)(ws + 503272);       // 40
    float* protoN  = ws + 503312;               // 3200

    k1_prep  <<<1,   256, 0, stream>>>(emb, mv, simSup);
    k2_simmem<<<512, 256, 0, stream>>>(mem, mv, simMem);
    k3_topk  <<<WAY, 1024, 0, stream>>>(simSup, simMem, topVals, topIdx);
    k4_proto <<<1,   1024, 0, stream>>>(emb, mem, topVals, topIdx, protoN);
    k5_logits<<<5,   32,  0, stream>>>(emb, protoN, out);
}